// EdgeMask_18150531792933
// MI455X (gfx1250) — compile-verified
//
#include <hip/hip_runtime.h>
#include <math.h>

typedef float v2f __attribute__((ext_vector_type(2)));
typedef float v8f __attribute__((ext_vector_type(8)));

#define EDGE_DIM 128
#define K_ITERS  5

// ---------------------------------------------------------------------------
// Stage 1: edge_weights[e] = sigmoid(dot(edge_attr[e], W) + b)
// One wave32 handles a 16-edge tile with V_WMMA_F32_16X16X4_F32 (exact f32).
// A (16x4): lane<16 -> (M=lane, K=k0..k0+1); lane>=16 -> (M=lane-16, K=k0+2..k0+3)
// B (4x16): column 0 = W chunk, rest zero  =>  D[:,0] = per-edge partial dots.
// D layout: lane 0 holds M=0..7 in c[0..7], lane 16 holds M=8..15.
// ---------------------------------------------------------------------------
__global__ __launch_bounds__(256) void k_edge_weights(
    const float* __restrict__ edge_attr, const float* __restrict__ W,
    const float* __restrict__ bptr, float* __restrict__ ew, int E)
{
    const int  lane = threadIdx.x & 31;
    const long wave = ((long)blockIdx.x * blockDim.x + threadIdx.x) >> 5;
    const long e0   = wave * 16;
    if (e0 >= E) return;

    const int half = lane >> 4;   // 0: K pair (0,1), 1: K pair (2,3)
    const int m    = lane & 15;   // row (edge) within tile
    const float bias = bptr[0];

    if (e0 + 16 <= (long)E) {
        const float* arow = edge_attr + (e0 + m) * EDGE_DIM + half * 2;
        v8f c = {0.f, 0.f, 0.f, 0.f, 0.f, 0.f, 0.f, 0.f};
#pragma unroll 4
        for (int k0 = 0; k0 < EDGE_DIM; k0 += 4) {
            v2f a = *(const v2f*)(arow + k0);
            v2f wv = *(const v2f*)(W + k0 + half * 2);
            v2f b;
            b.x = (m == 0) ? wv.x : 0.0f;
            b.y = (m == 0) ? wv.y : 0.0f;
            c = __builtin_amdgcn_wmma_f32_16x16x4_f32(
                    /*neg_a=*/false, a, /*neg_b=*/false, b,
                    /*c_mod=*/(short)0, c, /*reuse_a=*/false, /*reuse_b=*/false);
        }
        if (m == 0) {                      // lanes 0 and 16 own column 0 of D
            const long base = e0 + half * 8;
#pragma unroll
            for (int i = 0; i < 8; ++i) {
                float z = c[i] + bias;
                ew[base + i] = 1.0f / (1.0f + expf(-z));
            }
        }
    } else {
        // Ragged tail (<16 edges): scalar dot, one edge per lane. No WMMA here,
        // so EXEC divergence is fine.
        const long e = e0 + lane;
        if (e < (long)E) {
            const float* row = edge_attr + e * EDGE_DIM;
            float acc = 0.0f;
            for (int k = 0; k < EDGE_DIM; ++k) acc = fmaf(row[k], W[k], acc);
            ew[e] = 1.0f / (1.0f + expf(-(acc + bias)));
        }
    }
}

// h = relu(mask); x0 = h; deg = 1 (self loop weight)
__global__ __launch_bounds__(256) void k_init_nodes(
    const float* __restrict__ mask, float* __restrict__ h,
    float* __restrict__ x, float* __restrict__ deg, int N)
{
    int n = blockIdx.x * blockDim.x + threadIdx.x;
    if (n >= N) return;
    float v = mask[n];
    v = v > 0.0f ? v : 0.0f;
    h[n] = v;
    x[n] = v;
    deg[n] = 1.0f;
}

// deg[col[e]] += ew[e]
__global__ __launch_bounds__(256) void k_deg_scatter(
    float* __restrict__ deg, const int* __restrict__ col,
    const float* __restrict__ ew, int E)
{
    int e = blockIdx.x * blockDim.x + threadIdx.x;
    if (e >= E) return;
    atomicAdd(deg + col[e], ew[e]);
}

// in-place deg -> dinv
__global__ __launch_bounds__(256) void k_dinv(float* __restrict__ deg, int N)
{
    int n = blockIdx.x * blockDim.x + threadIdx.x;
    if (n >= N) return;
    float d = deg[n];
    deg[n] = d > 0.0f ? rsqrtf(d) : 0.0f;
}

// norm[e] = dinv[row]*ew*dinv[col]
__global__ __launch_bounds__(256) void k_norm(
    float* __restrict__ norm, const float* __restrict__ dinv,
    const int* __restrict__ row, const int* __restrict__ col,
    const float* __restrict__ ew, int E)
{
    int e = blockIdx.x * blockDim.x + threadIdx.x;
    if (e >= E) return;
    norm[e] = dinv[row[e]] * ew[e] * dinv[col[e]];
}

// y[n] = alpha*h[n] + (1-alpha)*dinv[n]^2*x[n]   (self-loop + teleport term)
__global__ __launch_bounds__(256) void k_self(
    float* __restrict__ y, const float* __restrict__ h,
    const float* __restrict__ x, const float* __restrict__ dinv,
    const float* __restrict__ alphap, int N)
{
    int n = blockIdx.x * blockDim.x + threadIdx.x;
    if (n >= N) return;
    float a = alphap[0];
    float di = dinv[n];
    y[n] = a * h[n] + (1.0f - a) * di * di * x[n];
}

// y[col[e]] += (1-alpha)*norm[e]*x[row[e]]
__global__ __launch_bounds__(256) void k_scatter(
    float* __restrict__ y, const float* __restrict__ x,
    const float* __restrict__ norm, const int* __restrict__ row,
    const int* __restrict__ col, const float* __restrict__ alphap, int E)
{
    int e = blockIdx.x * blockDim.x + threadIdx.x;
    if (e >= E) return;
    float a = alphap[0];
    atomicAdd(y + col[e], (1.0f - a) * norm[e] * x[row[e]]);
}

// fill[n] = tanh(x[n] - softplus(bias))
__global__ __launch_bounds__(256) void k_fill(
    float* __restrict__ out, const float* __restrict__ x,
    const float* __restrict__ biasp, int N)
{
    int n = blockIdx.x * blockDim.x + threadIdx.x;
    if (n >= N) return;
    float sp = log1pf(expf(biasp[0]));
    out[n] = tanhf(x[n] - sp);
}

extern "C" void kernel_launch(void* const* d_in, const int* in_sizes, int n_in,
                              void* d_out, int out_size, void* d_ws, size_t ws_size,
                              hipStream_t stream)
{
    const float* edge_attr = (const float*)d_in[0];
    const float* mask      = (const float*)d_in[1];
    const int*   eidx      = (const int*)d_in[2];   // int32 (JAX x64 off)
    const float* W         = (const float*)d_in[3];
    const float* b         = (const float*)d_in[4];
    const float* alpha     = (const float*)d_in[5];
    const float* bias      = (const float*)d_in[6];

    const int E = in_sizes[0] / EDGE_DIM;
    const int N = in_sizes[1];
    const int* row = eidx;
    const int* col = eidx + E;

    float* fill = (float*)d_out;       // [N]
    float* ew   = (float*)d_out + N;   // [E]

    // workspace carve-up (256B aligned)
    auto align256 = [](size_t v) { return (v + 255) & ~(size_t)255; };
    char* wsp = (char*)d_ws;
    float* h    = (float*)wsp;  wsp += align256((size_t)N * 4);
    float* x    = (float*)wsp;  wsp += align256((size_t)N * 4);
    float* y    = (float*)wsp;  wsp += align256((size_t)N * 4);
    float* dinv = (float*)wsp;  wsp += align256((size_t)N * 4);
    float* norm = (float*)wsp;  wsp += align256((size_t)E * 4);
    (void)ws_size; (void)n_in; (void)out_size;

    const int TB = 256;
    const int nblkN = (N + TB - 1) / TB;
    const int nblkE = (E + TB - 1) / TB;
    // edge-weight kernel: 16 edges per wave, 8 waves per block -> 128 edges/block
    const int nblkW = (E + 127) / 128;

    k_edge_weights<<<nblkW, TB, 0, stream>>>(edge_attr, W, b, ew, E);
    k_init_nodes<<<nblkN, TB, 0, stream>>>(mask, h, x, dinv /*deg*/, N);
    k_deg_scatter<<<nblkE, TB, 0, stream>>>(dinv /*deg*/, col, ew, E);
    k_dinv<<<nblkN, TB, 0, stream>>>(dinv, N);
    k_norm<<<nblkE, TB, 0, stream>>>(norm, dinv, row, col, ew, E);

    for (int it = 0; it < K_ITERS; ++it) {
        k_self<<<nblkN, TB, 0, stream>>>(y, h, x, dinv, alpha, N);
        k_scatter<<<nblkE, TB, 0, stream>>>(y, x, norm, row, col, alpha, E);
        float* t = x; x = y; y = t;
    }

    k_fill<<<nblkN, TB, 0, stream>>>(fill, x, bias, N);
}